// FFTLongConv_5033701671551
// MI455X (gfx1250) — compile-verified
//
#include <hip/hip_runtime.h>
#include <math.h>

typedef float v2f __attribute__((ext_vector_type(2)));
typedef float v8f __attribute__((ext_vector_type(8)));
typedef unsigned int u32x4 __attribute__((ext_vector_type(4)));
typedef int i32x8 __attribute__((ext_vector_type(8)));
typedef int i32x4 __attribute__((ext_vector_type(4)));

#define N_FFT    16384
#define L_SEQ    8192
#define NTHREADS 256
#define PI_F     3.14159265358979323846f
// byte offset of the staging area (== Bb) inside the dynamic-LDS block
#define STAGE_OFF (N_FFT * 8u)

// ---------------------------------------------------------------------------
// TDM: DMA one contiguous row of `nfloats` f32 from global into LDS.
// D# per CDNA5 ISA ch.8: group0 = {count=1, lds_addr, global_addr, type=2},
// group1 = {data_size=4B, tensor_dim0=nfloats, tensor_dim1=1, tile_dim0=nfloats,
//           tile_dim1=1, strides=nfloats}. Groups 2/3 unused (<=2D tensor).
// Tracked by TENSORcnt. EXEC is ignored; issue once per wave.
// (6-arg builtin form: g0, g1, g2, g3, g_extra, cpol)
// ---------------------------------------------------------------------------
__device__ __forceinline__ void tdm_load_row(unsigned lds_byte_off,
                                             const float* gptr, unsigned nfloats)
{
  const unsigned long long ga = (unsigned long long)(uintptr_t)gptr;
  u32x4 g0;
  g0[0] = 1u;                                                  // count=1 (valid)
  g0[1] = lds_byte_off;                                        // lds_addr
  g0[2] = (unsigned)(ga & 0xFFFFFFFFu);                        // global_addr[31:0]
  g0[3] = (unsigned)((ga >> 32) & 0x01FFFFFFu) | (2u << 30);   // addr[56:32] | type=2

  i32x8 g1;
  g1[0] = 0x00020000;                                          // data_size=2 -> 4 bytes
  g1[1] = (int)((nfloats & 0xFFFFu) << 16);                    // tensor_dim0[15:0]
  g1[2] = (int)(((nfloats >> 16) & 0xFFFFu) | (1u << 16));     // tensor_dim0[31:16], tensor_dim1=1
  g1[3] = (int)((nfloats & 0xFFFFu) << 16);                    // tile_dim0 = nfloats
  g1[4] = 1;                                                   // tile_dim1 = 1
  g1[5] = (int)nfloats;                                        // tensor_dim0_stride[31:0]
  g1[6] = (int)((nfloats & 0xFFFFu) << 16);                    // dim0_stride[47:32]=0, dim1_stride[15:0]
  g1[7] = 0;                                                   // tensor_dim1_stride[47:16]

  i32x4 z4; z4[0] = 0; z4[1] = 0; z4[2] = 0; z4[3] = 0;
  i32x8 z8; z8[0] = 0; z8[1] = 0; z8[2] = 0; z8[3] = 0;
  z8[4] = 0; z8[5] = 0; z8[6] = 0; z8[7] = 0;
  __builtin_amdgcn_tensor_load_to_lds(g0, g1, z4, z4, z8, 0);
}

// ---------------------------------------------------------------------------
// One radix-16 Stockham stage over 16384 complex values in LDS.
// 1024 butterflies -> 64 groups of 16 columns; each group is a complex
// 16x16x16 matmul done as 4 chained V_WMMA_F32_16X16X4_F32 per real product.
// Gather stride is l*m = 1024 (constant across all radix-16 stages).
// Twiddles via complex recurrence: 2 sincos + 7 cmuls per group (not 8 sincos).
// ---------------------------------------------------------------------------
__device__ __forceinline__ void radix16_stage(
    const float2* __restrict__ src, float2* __restrict__ dst,
    int m, int log2m, int wave, int half, int n16,
    const v2f* wre, const v2f* wim, const v2f* wimn)
{
#pragma unroll 1
  for (int it = 0; it < 8; ++it) {
    const int g = (wave << 3) + it;   // 64 groups, 8 per wave
    const int q = (g << 4) + n16;     // this lane's butterfly/column id
    v8f dre  = {0.f,0.f,0.f,0.f,0.f,0.f,0.f,0.f};
    v8f dim_ = {0.f,0.f,0.f,0.f,0.f,0.f,0.f,0.f};
#pragma unroll
    for (int c = 0; c < 4; ++c) {
      const int t0 = (c << 2) + (half << 1);       // K rows for this lane half
      const float2 z0 = src[q + (t0    ) * 1024];
      const float2 z1 = src[q + (t0 + 1) * 1024];
      v2f bre; bre[0] = z0.x; bre[1] = z1.x;
      v2f bim; bim[0] = z0.y; bim[1] = z1.y;
      // Dre = Wre*Bre - Wim*Bim ; Dim = Wre*Bim + Wim*Bre
      dre  = __builtin_amdgcn_wmma_f32_16x16x4_f32(false, wre[c],  false, bre, (short)0, dre,  false, false);
      dre  = __builtin_amdgcn_wmma_f32_16x16x4_f32(false, wimn[c], false, bim, (short)0, dre,  false, false);
      dim_ = __builtin_amdgcn_wmma_f32_16x16x4_f32(false, wre[c],  false, bim, (short)0, dim_, false, false);
      dim_ = __builtin_amdgcn_wmma_f32_16x16x4_f32(false, wim[c],  false, bre, (short)0, dim_, false, false);
    }
    const int j = q >> log2m;
    const int k = q & (m - 1);
    // twiddle w_{16l}^{j*s} = exp(-i*pi/8192 * j*m * s), s = v + 8*half
    const float th1 = -(PI_F / 8192.0f) * (float)(j * m);
    float c1, s1, cb, sb;
    __sincosf(th1, &s1, &c1);
    __sincosf(th1 * (float)(half << 3), &sb, &cb);
#pragma unroll
    for (int v = 0; v < 8; ++v) {
      const int s = v + (half << 3);               // output row held by this lane
      const float re = dre[v], im = dim_[v];
      float2 o;
      o.x = re * cb - im * sb;
      o.y = re * sb + im * cb;
      dst[k + (((j << 4) + s) << log2m)] = o;      // self-sorting scatter
      const float nc = cb * c1 - sb * s1;          // advance twiddle by th1
      const float ns = sb * c1 + cb * s1;
      cb = nc; sb = ns;
    }
  }
}

// Final radix-4 stage: l=1, m=4096, twiddle-free. Pure VALU.
__device__ __forceinline__ void radix4_stage(const float2* __restrict__ src,
                                             float2* __restrict__ dst, int tid)
{
#pragma unroll 1
  for (int i = 0; i < 16; ++i) {
    const int k = tid + (i << 8);
    const float2 c0 = src[k];
    const float2 c1 = src[k + 4096];
    const float2 c2 = src[k + 8192];
    const float2 c3 = src[k + 12288];
    const float e0x = c0.x + c2.x, e0y = c0.y + c2.y;
    const float e1x = c0.x - c2.x, e1y = c0.y - c2.y;
    const float o0x = c1.x + c3.x, o0y = c1.y + c3.y;
    const float o1x = c1.x - c3.x, o1y = c1.y - c3.y;
    float2 d0; d0.x = e0x + o0x; d0.y = e0y + o0y;
    float2 d2; d2.x = e0x - o0x; d2.y = e0y - o0y;
    float2 d1; d1.x = e1x + o1y; d1.y = e1y - o1x;   // e1 - i*o1
    float2 d3; d3.x = e1x - o1y; d3.y = e1y + o1x;   // e1 + i*o1
    dst[k]         = d0;
    dst[k + 4096]  = d1;
    dst[k + 8192]  = d2;
    dst[k + 12288] = d3;
  }
}

// Forward 16384-pt complex FFT, radices 16,16,16,4; result ends back in buf.
__device__ __forceinline__ void fft16k(float2* buf, float2* tmp,
                                       int tid, int wave, int half, int n16,
                                       const v2f* wre, const v2f* wim, const v2f* wimn)
{
  radix16_stage(buf, tmp, 1,   0, wave, half, n16, wre, wim, wimn); __syncthreads();
  radix16_stage(tmp, buf, 16,  4, wave, half, n16, wre, wim, wimn); __syncthreads();
  radix16_stage(buf, tmp, 256, 8, wave, half, n16, wre, wim, wimn); __syncthreads();
  radix4_stage (tmp, buf, tid);                                     __syncthreads();
}

extern "C" __global__ __launch_bounds__(NTHREADS, 1)
void fftconv16k_kernel(const float* __restrict__ x,
                       const float* __restrict__ h,
                       float* __restrict__ y)
{
  extern __shared__ float2 smem[];          // 2 * 16384 complex = 256 KB
  float2* A  = smem;
  float2* Bb = smem + N_FFT;

  const int tid  = threadIdx.x;
  const int lane = tid & 31;
  const int wave = tid >> 5;
  const int half = lane >> 4;
  const int n16  = lane & 15;
  const int batch = blockIdx.x;

  const float* xr = x + (size_t)batch * L_SEQ;
  const float* hr = h + (size_t)batch * L_SEQ;

  // Kick off the TDM copies first: x -> staging[0:8192), h -> staging[8192:16384)
  // (staging aliases Bb, which is dead until FFT stage 0 writes it).
  if (wave == 0) {
    tdm_load_row(STAGE_OFF,                xr, L_SEQ);
    tdm_load_row(STAGE_OFF + L_SEQ * 4u,   hr, L_SEQ);
  }

  // While the DMA runs: per-lane DFT16 A-operand chunks, W[m,t]=exp(-2pi*i*m*t/16),
  // laid out per ISA 16x4 f32 A-matrix layout (M = lane&15, K split by half).
  v2f wre[4], wim[4], wimn[4];
#pragma unroll
  for (int c = 0; c < 4; ++c) {
    const int t0 = (c << 2) + (half << 1);
    float s0, c0f, s1, c1f;
    __sincosf(-(PI_F / 8.0f) * (float)(n16 * t0),       &s0, &c0f);
    __sincosf(-(PI_F / 8.0f) * (float)(n16 * (t0 + 1)), &s1, &c1f);
    wre[c][0]  = c0f;  wre[c][1]  = c1f;
    wim[c][0]  = s0;   wim[c][1]  = s1;
    wimn[c][0] = -s0;  wimn[c][1] = -s1;
  }

  // Zero-fill the padded upper half of A (independent of staging).
  float2 zz; zz.x = 0.f; zz.y = 0.f;
  for (int t = L_SEQ + tid; t < N_FFT; t += NTHREADS) A[t] = zz;

  if (wave == 0) __builtin_amdgcn_s_wait_tensorcnt(0);
  __syncthreads();

  // Pack z = x + i*h into A from the staged rows (LDS b128 reads).
  const float4* xs4 = (const float4*)(Bb);
  const float4* hs4 = xs4 + (L_SEQ / 4);
  for (int i = tid; i < L_SEQ / 4; i += NTHREADS) {
    const float4 xv = xs4[i];
    const float4 hv = hs4[i];
    const int t = i << 2;
    float2 p;
    p.x = xv.x; p.y = hv.x; A[t + 0] = p;
    p.x = xv.y; p.y = hv.y; A[t + 1] = p;
    p.x = xv.z; p.y = hv.z; A[t + 2] = p;
    p.x = xv.w; p.y = hv.w; A[t + 3] = p;
  }
  __syncthreads();

  // Forward FFT of packed z: result Z in A.
  fft16k(A, Bb, tid, wave, half, n16, wre, wim, wimn);

  // Split spectra (x,h real): X = (Z_k + conj(Z_{n-k}))/2, H = (Z_k - conj(Z_{n-k}))/(2i).
  // S = X*H; store P = conj(S)/n^2 so the same forward FFT computes the inverse:
  // y_t = Re( fft(P)_t ).
  const float invn2 = 1.0f / ((float)N_FFT * (float)N_FFT);
#pragma unroll 1
  for (int i = 0; i < N_FFT / NTHREADS; ++i) {
    const int k  = tid + (i << 8);
    const int kr = (N_FFT - k) & (N_FFT - 1);
    const float2 u = A[k];
    const float2 w = A[kr];
    const float vx = w.x, vy = -w.y;                 // conj(Z_{n-k})
    const float Xx = 0.5f * (u.x + vx), Xy = 0.5f * (u.y + vy);
    const float Wx = u.x - vx,          Wy = u.y - vy;
    const float Hx = 0.5f * Wy,         Hy = -0.5f * Wx;   // W/(2i)
    const float Sx = Xx * Hx - Xy * Hy;
    const float Sy = Xx * Hy + Xy * Hx;
    float2 p; p.x = Sx * invn2; p.y = -Sy * invn2;   // conj(S)/n^2
    Bb[k] = p;
  }
  __syncthreads();

  // "Inverse" via forward FFT of conjugated spectrum: result in Bb.
  fft16k(Bb, A, tid, wave, half, n16, wre, wim, wimn);

  // y = Re(result)[0:L]
  float* yr = y + (size_t)batch * L_SEQ;
  for (int t = tid; t < L_SEQ; t += NTHREADS) yr[t] = Bb[t].x;
}

extern "C" void kernel_launch(void* const* d_in, const int* in_sizes, int n_in,
                              void* d_out, int out_size, void* d_ws, size_t ws_size,
                              hipStream_t stream)
{
  (void)n_in; (void)out_size; (void)d_ws; (void)ws_size;
  const float* x = (const float*)d_in[0];
  const float* h = (const float*)d_in[1];
  float* y = (float*)d_out;

  const int B = in_sizes[0] / L_SEQ;
  const size_t smem_bytes = (size_t)2 * N_FFT * sizeof(float2);   // 256 KB / WGP

  (void)hipFuncSetAttribute((const void*)fftconv16k_kernel,
                            hipFuncAttributeMaxDynamicSharedMemorySize,
                            (int)smem_bytes);

  fftconv16k_kernel<<<B, NTHREADS, smem_bytes, stream>>>(x, h, y);
}